// EdgeAttention_23991687315683
// MI455X (gfx1250) — compile-verified
//
#include <hip/hip_runtime.h>
#include <stdint.h>

// EdgeAttention (triangle-attention style) for MI455X / gfx1250.
// B=2, N=28, D=128, H=8, DK=16. All GEMM stages use V_WMMA_F32_16X16X4_F32
// (exact fp32 FMA semantics, wave32 matrix pipe).

typedef __attribute__((ext_vector_type(2))) float v2f;
typedef __attribute__((ext_vector_type(8))) float v8f;

#define NB 2
#define NN 28
#define ND 128
#define NH 8
#define NDK 16
#define NROWS (NB * NN * NN)          // 1568 rows, 98 tiles of 16
#define SC_ELEMS (NB * NH * NN * NN * NN)  // 351232
#define PROJ_ELEMS (NROWS * ND)            // 200704

static __device__ __forceinline__ v8f wmma_f32_k4(v2f a, v2f b, v8f c) {
  // D = A(16x4,f32) x B(4x16,f32) + C(16x16,f32)
  return __builtin_amdgcn_wmma_f32_16x16x4_f32(
      /*neg_a=*/false, a, /*neg_b=*/false, b,
      /*c_mod=*/(short)0, c, /*reuse_a=*/false, /*reuse_b=*/false);
}

// ---------------------------------------------------------------------------
// Shared 16x16-tile GEMM body: Y[1568x128] = X[1568x128] @ W[128x128]^T
// One wave per 16x16 output tile. blockIdx.x = row tile (0..97),
// wave-in-block = column tile (0..7). 32 chained k4 WMMAs per tile.
// ---------------------------------------------------------------------------
static __device__ __forceinline__ void gemm_rowtile(const float* __restrict__ X,
                                                    const float* __restrict__ W,
                                                    float* __restrict__ Y) {
  const int lane = threadIdx.x & 31;
  const int tn   = threadIdx.x >> 5;   // 0..7  (column tile)
  const int tm   = blockIdx.x;         // 0..97 (row tile)
  const int l    = lane & 15;
  const int hi   = lane >> 4;

  const float* xrow = X + (size_t)(tm * 16 + l) * ND;  // A row M=l
  const float* wrow = W + (size_t)(tn * 16 + l) * ND;  // B col N=l (B[k][n]=W[n][k])

  v8f acc = {};
#pragma unroll
  for (int k0 = 0; k0 < ND; k0 += 4) {
    const int k = k0 + 2 * hi;
    v2f a = *(const v2f*)(xrow + k);
    v2f b = *(const v2f*)(wrow + k);
    acc = wmma_f32_k4(a, b, acc);
  }

  const int col = tn * 16 + l;
#pragma unroll
  for (int v = 0; v < 8; ++v) {
    const int row = tm * 16 + v + 8 * hi;
    Y[(size_t)row * ND + col] = acc[v];
  }
}

// Fused four projections: blockIdx.y selects {LK,RK,LV,RV}.
__global__ __launch_bounds__(256) void proj_kernel(
    const float* __restrict__ key, const float* __restrict__ value,
    const float* __restrict__ Wlk, const float* __restrict__ Wrk,
    const float* __restrict__ Wlv, const float* __restrict__ Wrv,
    float* __restrict__ LK, float* __restrict__ RK,
    float* __restrict__ LV, float* __restrict__ RV) {
  const float* X;
  const float* W;
  float* Y;
  switch (blockIdx.y) {
    case 0:  X = key;   W = Wlk; Y = LK; break;
    case 1:  X = key;   W = Wrk; Y = RK; break;
    case 2:  X = value; W = Wlv; Y = LV; break;
    default: X = value; W = Wrv; Y = RV; break;
  }
  gemm_rowtile(X, W, Y);
}

// Output projection: out = XI @ Wout^T
__global__ __launch_bounds__(256) void outproj_kernel(
    const float* __restrict__ XI, const float* __restrict__ Wout,
    float* __restrict__ out) {
  gemm_rowtile(XI, Wout, out);
}

// ---------------------------------------------------------------------------
// Scores: for each (b,h,a) compute S[x,y] = LK[b,x,a,h,:] . RK[b,a,y,h,:]
// as a 28x28x16 WMMA GEMM (4 waves = 2x2 tiles of 16x16, 4 k4-steps each).
// Fuses the 1/sqrt(DK) scale and the -1e9 mask into the store.
// SC layout: [b][h][x][a][y]
// ---------------------------------------------------------------------------
__global__ __launch_bounds__(128) void scores_kernel(
    const float* __restrict__ LK, const float* __restrict__ RK,
    const unsigned char* __restrict__ mask, float* __restrict__ SC) {
  const int job = blockIdx.x;        // (b*8 + h)*28 + a, 448 jobs
  const int a = job % NN;
  const int t = job / NN;
  const int h = t % NH;
  const int b = t / NH;

  const int wave = threadIdx.x >> 5;  // 0..3
  const int lane = threadIdx.x & 31;
  const int tmn  = wave >> 1;         // row tile 0..1
  const int tnn  = wave & 1;          // col tile 0..1
  const int l    = lane & 15;
  const int hi   = lane >> 4;

  const int x  = tmn * 16 + l;               // A row (may exceed 27)
  const int y  = tnn * 16 + l;               // B col (may exceed 27)
  const int xc = x < NN ? x : NN - 1;        // clamp keeps EXEC all-ones
  const int yc = y < NN ? y : NN - 1;

  const float* arow = LK + ((size_t)((b * NN + xc) * NN + a)) * ND + h * NDK;
  const float* brow = RK + ((size_t)((b * NN + a) * NN + yc)) * ND + h * NDK;

  v8f acc = {};
#pragma unroll
  for (int k0 = 0; k0 < NDK; k0 += 4) {
    const int k = k0 + 2 * hi;
    v2f av = *(const v2f*)(arow + k);
    v2f bv = *(const v2f*)(brow + k);
    acc = wmma_f32_k4(av, bv, acc);
  }

#pragma unroll
  for (int v = 0; v < 8; ++v) {
    const int xx = tmn * 16 + v + 8 * hi;
    if (xx < NN && y < NN) {
      const bool mk =
          mask[((size_t)(b * NN + xx) * NN + a) * NN + y] != 0;
      const float s = mk ? -1e9f : 0.25f * acc[v];  // 1/sqrt(16) = 0.25
      SC[((((size_t)(b * NH + h) * NN + xx) * NN + a) * NN) + y] = s;
    }
  }
}

// ---------------------------------------------------------------------------
// Softmax over 'a' (in place). One thread per (b,h,x,y) = 12544 threads.
// ---------------------------------------------------------------------------
__global__ __launch_bounds__(256) void softmax_kernel(float* __restrict__ SC) {
  const int t = blockIdx.x * 256 + threadIdx.x;  // exactly 12544
  const int y = t % NN;
  int r = t / NN;
  const int x = r % NN;
  r /= NN;
  const int h = r % NH;
  const int b = r / NH;

  float* base = SC + (((size_t)(b * NH + h) * NN + x) * NN * NN) + y;  // +a*28

  float vals[NN];
  float mx = -3.402823466e38f;
#pragma unroll
  for (int a = 0; a < NN; ++a) {
    vals[a] = base[a * NN];
    mx = fmaxf(mx, vals[a]);
  }
  float sum = 0.f;
#pragma unroll
  for (int a = 0; a < NN; ++a) {
    const float e = __expf(vals[a] - mx);
    vals[a] = e;
    sum += e;
  }
  const float inv = 1.0f / sum;
#pragma unroll
  for (int a = 0; a < NN; ++a) base[a * NN] = vals[a] * inv;
}

// ---------------------------------------------------------------------------
// Trilinear combine: XI[b,x,y,h,d] = sum_a att[b,h,x,a,y]*LV[b,x,a,h,d]*RV[b,a,y,h,d]
// Elementwise in d -> VALU FMA (only ~11 MFLOP). One thread per output elem,
// d fastest so LV/RV loads are contiguous 64B per 16-lane group.
// ---------------------------------------------------------------------------
__global__ __launch_bounds__(256) void combine_kernel(
    const float* __restrict__ LV, const float* __restrict__ RV,
    const float* __restrict__ SC, float* __restrict__ XI) {
  const int t = blockIdx.x * 256 + threadIdx.x;  // exactly 200704
  const int d = t & 15;
  int r = t >> 4;
  const int h = r & 7;
  r >>= 3;
  const int y = r % NN;
  r /= NN;
  const int x = r % NN;
  const int b = r / NN;

  const float* att = SC + (((size_t)(b * NH + h) * NN + x) * NN * NN) + y;  // +a*28
  const float* lv  = LV + ((size_t)(b * NN + x) * NN) * ND + h * NDK + d;   // +a*128
  const float* rv  = RV + ((size_t)b * NN * NN) * ND + (size_t)y * ND + h * NDK + d;  // +a*28*128

  float acc = 0.f;
#pragma unroll 4
  for (int a = 0; a < NN; ++a) {
    acc = fmaf(att[a * NN] * lv[(size_t)a * ND], rv[(size_t)a * NN * ND], acc);
  }
  XI[((size_t)(b * NN + x) * NN + y) * ND + h * NDK + d] = acc;
}

// ---------------------------------------------------------------------------
extern "C" void kernel_launch(void* const* d_in, const int* in_sizes, int n_in,
                              void* d_out, int out_size, void* d_ws, size_t ws_size,
                              hipStream_t stream) {
  // inputs: 0=query(unused) 1=key 2=value 3=mask(bool,1B) 4=Wlk 5=Wrk 6=Wlv
  //         7=Wrv 8=Wq(unused) 9=Wout
  const float* key   = (const float*)d_in[1];
  const float* value = (const float*)d_in[2];
  const unsigned char* mask = (const unsigned char*)d_in[3];
  const float* Wlk  = (const float*)d_in[4];
  const float* Wrk  = (const float*)d_in[5];
  const float* Wlv  = (const float*)d_in[6];
  const float* Wrv  = (const float*)d_in[7];
  const float* Wout = (const float*)d_in[9];
  float* out = (float*)d_out;

  float* ws = (float*)d_ws;
  float* LK = ws;                    // 200704 floats each
  float* RK = LK + PROJ_ELEMS;
  float* LV = RK + PROJ_ELEMS;
  float* RV = LV + PROJ_ELEMS;
  float* SC = RV + PROJ_ELEMS;       // 351232 floats
  float* XI = SC + SC_ELEMS;         // 200704 floats

  // 1) four projections, WMMA: grid (98 row-tiles, 4 matrices), 8 waves/block
  proj_kernel<<<dim3(98, 4), 256, 0, stream>>>(key, value, Wlk, Wrk, Wlv, Wrv,
                                               LK, RK, LV, RV);
  // 2) 448 small 28x28x16 GEMMs + scale + mask, WMMA
  scores_kernel<<<dim3(NB * NH * NN), 128, 0, stream>>>(LK, RK, mask, SC);
  // 3) softmax over 'a' (12544 threads)
  softmax_kernel<<<dim3(49), 256, 0, stream>>>(SC);
  // 4) trilinear combine (200704 threads)
  combine_kernel<<<dim3(784), 256, 0, stream>>>(LV, RV, SC, XI);
  // 5) output projection, WMMA
  outproj_kernel<<<dim3(98), 256, 0, stream>>>(XI, Wout, out);
}